// SAGEEdgeClassifier_85126251807612
// MI455X (gfx1250) — compile-verified
//
#include <hip/hip_runtime.h>
#include <hip/hip_bf16.h>

typedef __attribute__((ext_vector_type(2))) float v2f;
typedef __attribute__((ext_vector_type(8))) float v8f;

static __device__ __forceinline__ v8f wmma4(v2f a, v2f b, v8f c) {
    // D = A(16x4) * B(4x16) + C, full fp32
    return __builtin_amdgcn_wmma_f32_16x16x4_f32(
        /*neg_a=*/false, a, /*neg_b=*/false, b,
        /*c_mod=*/(short)0, c, /*reuse_a=*/false, /*reuse_b=*/false);
}

// ---------------- utility kernels ----------------

__global__ void zero_f32(float* __restrict__ p, long long n) {
    long long i = (long long)blockIdx.x * blockDim.x + threadIdx.x;
    if (i < n) p[i] = 0.0f;
}

__global__ void degree_k(const int* __restrict__ dst, float* __restrict__ cnt, int E) {
    int i = blockIdx.x * blockDim.x + threadIdx.x;
    if (i < E) atomicAdd(&cnt[dst[i]], 1.0f);
}

__global__ void invcnt_k(float* __restrict__ cnt, int N) {
    int i = blockIdx.x * blockDim.x + threadIdx.x;
    if (i < N) cnt[i] = 1.0f / fmaxf(cnt[i], 1.0f);
}

// scatter-add: agg[dst[e]][f] += feat[src[e]][f], F = 1<<logF
__global__ void scatter_add_k(const float* __restrict__ feat,
                              const int* __restrict__ src,
                              const int* __restrict__ dst,
                              float* __restrict__ agg,
                              long long total, int logF) {
    long long i = (long long)blockIdx.x * blockDim.x + threadIdx.x;
    if (i >= total) return;
    long long e = i >> logF;
    int f = (int)(i & ((1 << logF) - 1));
    int s = src[e], d = dst[e];
    atomicAdd(&agg[(size_t)d * (1 << logF) + f],
              feat[(size_t)s * (1 << logF) + f]);
}

// ---------------- SAGE layer GEMM ----------------
// out[row, 0..127] = act( (agg[row]*inv[row]) @ Wl + bias + xin[row] @ Wr )
// Block = 256 threads = 8 waves. Block handles 16 rows; wave w handles cols [16w,16w+16).
template <int K, bool RELU, bool INPLACE>
__global__ __launch_bounds__(256) void sage_gemm_k(
    const float* __restrict__ agg, const float* __restrict__ inv,
    const float* __restrict__ xin,
    const float* __restrict__ Wl, const float* __restrict__ Wr,
    const float* __restrict__ bias, float* __restrict__ out, int N) {
    const int t = threadIdx.x;
    const int lane = t & 31;
    const int wid = t >> 5;          // 0..7 -> output col tile
    const int half = lane >> 4;      // 0 or 1
    const int l = lane & 15;
    const int row0 = blockIdx.x * 16;
    const int arow = min(row0 + l, N - 1);   // A-matrix row for this lane
    const int n = wid * 16 + l;              // output column (0..127)

    const float invr = inv[arow];
    const float* aggRow = agg + (size_t)arow * K;
    const float* xRow   = xin + (size_t)arow * K;

    v8f c = {};
    #pragma unroll
    for (int k = 0; k < K; k += 4) {
        const int kk = k + 2 * half;         // this lane's K pair: kk, kk+1
        v2f al = { aggRow[kk] * invr, aggRow[kk + 1] * invr };
        v2f ar = { xRow[kk], xRow[kk + 1] };
        v2f bl = { Wl[(size_t)kk * 128 + n], Wl[(size_t)(kk + 1) * 128 + n] };
        v2f br = { Wr[(size_t)kk * 128 + n], Wr[(size_t)(kk + 1) * 128 + n] };
        c = wmma4(al, bl, c);
        c = wmma4(ar, br, c);
    }

    if (INPLACE) __syncthreads();  // all reads of this block's rows done before in-place stores

    const float b = bias[n];
    #pragma unroll
    for (int v = 0; v < 8; ++v) {
        const int r = row0 + v + 8 * half;   // D-matrix row
        if (r < N) {
            float val = c[v] + b;
            if (RELU) val = fmaxf(val, 0.0f);
            out[(size_t)r * 128 + n] = val;
        }
    }
}

// ---------------- edge classifier ----------------
// 16 flow-edges per block. feats = [h[fsrc](128) | h[fdst](128) | eattr(32)] (K=288)
// hidden = relu(feats @ W1 + b1); out = hidden @ W2 + b2
__global__ __launch_bounds__(256) void edge_mlp_k(
    const float* __restrict__ h, const int* __restrict__ fsrc,
    const int* __restrict__ fdst, const float* __restrict__ eattr,
    const float* __restrict__ W1, const float* __restrict__ b1,
    const float* __restrict__ W2, const float* __restrict__ b2,
    float* __restrict__ out, int Ef) {
    __shared__ float acc[16];
    __shared__ int ssrc[16];
    __shared__ int sdst[16];

    const int e0 = blockIdx.x * 16;
    const int t = threadIdx.x;
    if (t < 16) {
        int e = min(e0 + t, Ef - 1);
        acc[t] = 0.0f;
        ssrc[t] = fsrc[e];
        sdst[t] = fdst[e];
    }
    __syncthreads();

    const int lane = t & 31;
    const int wid = t >> 5;
    const int half = lane >> 4;
    const int l = lane & 15;
    const int n = wid * 16 + l;                       // hidden column (0..127)
    const int erow = min(e0 + l, Ef - 1);

    const float* hs = h + (size_t)ssrc[l] * 128;
    const float* hd = h + (size_t)sdst[l] * 128;
    const float* ea = eattr + (size_t)erow * 32;

    v8f c = {};
    #pragma unroll
    for (int k = 0; k < 288; k += 4) {
        const int kk = k + 2 * half;
        v2f a;
        if (k < 128)        a = { hs[kk],       hs[kk + 1] };
        else if (k < 256)   a = { hd[kk - 128], hd[kk - 127] };
        else                a = { ea[kk - 256], ea[kk - 255] };
        v2f b = { W1[(size_t)kk * 128 + n], W1[(size_t)(kk + 1) * 128 + n] };
        c = wmma4(a, b, c);
    }

    const float bb = b1[n];
    const float w2 = W2[n];
    #pragma unroll
    for (int v = 0; v < 8; ++v) {
        float val = fmaxf(c[v] + bb, 0.0f) * w2;     // hidden row = v + 8*half
        atomicAdd(&acc[v + 8 * half], val);          // LDS ds_add_f32 reduce across 128 cols
    }
    __syncthreads();
    if (t < 16 && (e0 + t) < Ef) out[e0 + t] = acc[t] + b2[0];
}

// ---------------- launcher ----------------

extern "C" void kernel_launch(void* const* d_in, const int* in_sizes, int n_in,
                              void* d_out, int out_size, void* d_ws, size_t ws_size,
                              hipStream_t stream) {
    const float* x     = (const float*)d_in[0];
    const int*   ei    = (const int*)d_in[1];
    const int*   fei   = (const int*)d_in[2];
    const float* eattr = (const float*)d_in[3];
    const float* Wl0   = (const float*)d_in[4];
    const float* bl0   = (const float*)d_in[5];
    const float* Wr0   = (const float*)d_in[6];
    const float* Wl1   = (const float*)d_in[7];
    const float* bl1   = (const float*)d_in[8];
    const float* Wr1   = (const float*)d_in[9];
    const float* W1    = (const float*)d_in[10];
    const float* b1    = (const float*)d_in[11];
    const float* W2    = (const float*)d_in[12];
    const float* b2    = (const float*)d_in[13];

    const int N  = in_sizes[0] / 64;   // 100000
    const int E  = in_sizes[1] / 2;    // 800000
    const int Ef = in_sizes[2] / 2;    // 500000

    const int* src  = ei;
    const int* dst  = ei + E;
    const int* fsrc = fei;
    const int* fdst = fei + Ef;

    // workspace layout (floats): cnt[N] (becomes inv-degree) | agg[N*128] | h[N*128]
    float* cnt = (float*)d_ws;
    float* agg = cnt + (((size_t)N + 255) & ~(size_t)255);
    float* h   = agg + (size_t)N * 128;

    const int T = 256;
    dim3 blk(T);

    // inverse mean-degree (shared by both SAGE layers)
    zero_f32<<<(N + T - 1) / T, blk, 0, stream>>>(cnt, N);
    degree_k<<<(E + T - 1) / T, blk, 0, stream>>>(dst, cnt, E);
    invcnt_k<<<(N + T - 1) / T, blk, 0, stream>>>(cnt, N);

    // --- layer 0: h = relu(mean(x[src]->dst) @ Wl0 + bl0 + x @ Wr0) ---
    {
        long long tot = (long long)N * 64;
        zero_f32<<<(unsigned)((tot + T - 1) / T), blk, 0, stream>>>(agg, tot);
        long long work = (long long)E * 64;
        scatter_add_k<<<(unsigned)((work + T - 1) / T), blk, 0, stream>>>(
            x, src, dst, agg, work, 6);
        sage_gemm_k<64, true, false><<<(N + 15) / 16, blk, 0, stream>>>(
            agg, cnt, x, Wl0, Wr0, bl0, h, N);
    }

    // --- layer 1: h = mean(h[src]->dst) @ Wl1 + bl1 + h @ Wr1 (in place) ---
    {
        long long tot = (long long)N * 128;
        zero_f32<<<(unsigned)((tot + T - 1) / T), blk, 0, stream>>>(agg, tot);
        long long work = (long long)E * 128;
        scatter_add_k<<<(unsigned)((work + T - 1) / T), blk, 0, stream>>>(
            h, src, dst, agg, work, 7);
        sage_gemm_k<128, false, true><<<(N + 15) / 16, blk, 0, stream>>>(
            agg, cnt, h, Wl1, Wr1, bl1, h, N);
    }

    // --- edge classifier ---
    edge_mlp_k<<<(Ef + 15) / 16, blk, 0, stream>>>(
        h, fsrc, fdst, eattr, W1, b1, W2, b2, (float*)d_out, Ef);
}